// ConstraintWholePoseScoringModule_49761491091730
// MI455X (gfx1250) — compile-verified
//
#include <hip/hip_runtime.h>
#include <stdint.h>

#define NPOSES  8
#define NBLOCKS 512
#define APB     16
#define NC      2000000
#define NPAIR   200000
#define NB2     (NBLOCKS * NBLOCKS)
#define FLATN   (NPOSES * NB2)      /* 2,097,152 floats = 8 MB */
#define EPSF    1e-7f

typedef int   v4i __attribute__((ext_vector_type(4)));
typedef float v2f __attribute__((ext_vector_type(2)));
typedef float v4f __attribute__((ext_vector_type(4)));

// Native no-return fp32 atomic add at device scope (L2 atomic unit).
__device__ __forceinline__ void atomic_add_f32_dev(float* addr, float v) {
    asm volatile("global_atomic_add_f32 %0, %1, off scope:SCOPE_DEV"
                 :: "v"((uint64_t)(uintptr_t)addr), "v"(v)
                 : "memory");
}

__global__ void zero_flat_kernel(v4f* __restrict__ flat4) {
    int t = blockIdx.x * blockDim.x + threadIdx.x;
    if (t < FLATN / 4) {
        v4f z = {0.f, 0.f, 0.f, 0.f};
        flat4[t] = z;
    }
}

__global__ void scatter_scores_kernel(const float* __restrict__ coords,
                                      const float* __restrict__ params,
                                      const int*   __restrict__ bco,
                                      const int*   __restrict__ pose,
                                      const int*   __restrict__ res,
                                      const int*   __restrict__ atom,
                                      const int*   __restrict__ types,
                                      float*       __restrict__ flat) {
    int t = blockIdx.x * blockDim.x + threadIdx.x;
    if (t >= NC) return;

    // Read-once streams: non-temporal so L2 keeps coords + accumulator hot.
    v4i r  = __builtin_nontemporal_load((const v4i*)res  + t);
    v4i a  = __builtin_nontemporal_load((const v4i*)atom + t);
    int p  = __builtin_nontemporal_load(pose + 4 * t);     // (NC,4) broadcast, col 0
    int ty = __builtin_nontemporal_load(types + t);
    v2f pr = __builtin_nontemporal_load((const v2f*)params + t);
    float x0 = pr.x, sd = pr.y;

    int rr[4] = {r.x, r.y, r.z, r.w};
    int aa[4] = {a.x, a.y, a.z, a.w};

    float ax[4], ay[4], az[4];
    const float* cbase = coords + (size_t)p * (NBLOCKS * APB * 3);
#pragma unroll
    for (int k = 0; k < 4; ++k) {
        int g = bco[p * NBLOCKS + rr[k]] + aa[k];   // small table, L2/WGP$ hot
        const float* c = cbase + (size_t)g * 3;
        ax[k] = c[0]; ay[k] = c[1]; az[k] = c[2];
    }

    float score;
    if (ty == 0) {
        // harmonic distance: ((|a0-a3| - x0)/sd)^2
        float dx = ax[0] - ax[3], dy = ay[0] - ay[3], dz = az[0] - az[3];
        float d = sqrtf(dx * dx + dy * dy + dz * dz);
        float u = (d - x0) / sd;
        score = u * u;
    } else if (ty == 1) {
        // harmonic angle
        float v1x = ax[0] - ax[1], v1y = ay[0] - ay[1], v1z = az[0] - az[1];
        float v2x = ax[2] - ax[1], v2y = ay[2] - ay[1], v2z = az[2] - az[1];
        float dot = v1x * v2x + v1y * v2y + v1z * v2z;
        float n1  = sqrtf(v1x * v1x + v1y * v1y + v1z * v1z);
        float n2  = sqrtf(v2x * v2x + v2y * v2y + v2z * v2z);
        float c   = dot / (n1 * n2 + EPSF);
        c = fminf(fmaxf(c, -1.0f + EPSF), 1.0f - EPSF);
        float ang = acosf(c);
        float u = (ang - x0) / sd;
        score = u * u;
    } else {
        // circular dihedral
        float b0x = ax[1] - ax[0], b0y = ay[1] - ay[0], b0z = az[1] - az[0];
        float b1x = ax[2] - ax[1], b1y = ay[2] - ay[1], b1z = az[2] - az[1];
        float b2x = ax[3] - ax[2], b2y = ay[3] - ay[2], b2z = az[3] - az[2];
        // n1 = b0 x b1 ; n2 = b1 x b2
        float n1x = b0y * b1z - b0z * b1y;
        float n1y = b0z * b1x - b0x * b1z;
        float n1z = b0x * b1y - b0y * b1x;
        float n2x = b1y * b2z - b1z * b2y;
        float n2y = b1z * b2x - b1x * b2z;
        float n2z = b1x * b2y - b1y * b2x;
        float b1l = sqrtf(b1x * b1x + b1y * b1y + b1z * b1z);
        float inv = 1.0f / (b1l + EPSF);
        float bnx = b1x * inv, bny = b1y * inv, bnz = b1z * inv;
        // m = n1 x b1n
        float mx = n1y * bnz - n1z * bny;
        float my = n1z * bnx - n1x * bnz;
        float mz = n1x * bny - n1y * bnx;
        float yv = mx * n2x + my * n2y + mz * n2z;
        float xv = n1x * n2x + n1y * n2y + n1z * n2z;
        float dih  = atan2f(yv, xv);
        float diff = dih - x0;
        float w = atan2f(sinf(diff), cosf(diff));
        float u = w / sd;
        score = u * u;
    }

    // Canonical upper-triangular accumulation: one atomic per constraint.
    int r0 = rr[0], r3 = rr[3];
    int lo = r0 < r3 ? r0 : r3;
    int hi = r0 < r3 ? r3 : r0;
    int idx = p * NB2 + lo * NBLOCKS + hi;
    atomic_add_f32_dev(flat + idx, score);
}

__global__ void gather_out_kernel(const int*   __restrict__ di,
                                  const float* __restrict__ flat,
                                  float*       __restrict__ out) {
    __shared__ int sp[256];
    __shared__ int si[256];
    __shared__ int sj[256];
    int t = blockIdx.x * 256 + threadIdx.x;

    if (t < NPAIR) {
        // Async global->LDS staging of the three index rows (ASYNCcnt path).
        uint32_t l0 = (uint32_t)(uintptr_t)(&sp[threadIdx.x]);
        uint32_t l1 = (uint32_t)(uintptr_t)(&si[threadIdx.x]);
        uint32_t l2 = (uint32_t)(uintptr_t)(&sj[threadIdx.x]);
        uint64_t g0 = (uint64_t)(uintptr_t)(di + t);
        uint64_t g1 = (uint64_t)(uintptr_t)(di + NPAIR + t);
        uint64_t g2 = (uint64_t)(uintptr_t)(di + 2 * NPAIR + t);
        asm volatile("global_load_async_to_lds_b32 %0, %1, off" :: "v"(l0), "v"(g0) : "memory");
        asm volatile("global_load_async_to_lds_b32 %0, %1, off" :: "v"(l1), "v"(g1) : "memory");
        asm volatile("global_load_async_to_lds_b32 %0, %1, off" :: "v"(l2), "v"(g2) : "memory");
    }
    asm volatile("s_wait_asynccnt 0" ::: "memory");
    __syncthreads();

    if (t < NPAIR) {
        int p = sp[threadIdx.x];
        int i = si[threadIdx.x];
        int j = sj[threadIdx.x];
        float v = flat[(size_t)p * NB2 + (size_t)i * NBLOCKS + j];
        __builtin_nontemporal_store(v, out + t);
    }
}

extern "C" void kernel_launch(void* const* d_in, const int* in_sizes, int n_in,
                              void* d_out, int out_size, void* d_ws, size_t ws_size,
                              hipStream_t stream) {
    const float* coords = (const float*)d_in[0];
    const float* params = (const float*)d_in[1];
    const int*   bco    = (const int*)d_in[2];
    const int*   pose   = (const int*)d_in[3];
    const int*   res    = (const int*)d_in[4];
    const int*   atom   = (const int*)d_in[5];
    const int*   types  = (const int*)d_in[6];
    const int*   di     = (const int*)d_in[7];
    float* flat = (float*)d_ws;     // 8 MB accumulator
    float* out  = (float*)d_out;

    zero_flat_kernel<<<(FLATN / 4 + 255) / 256, 256, 0, stream>>>((v4f*)flat);
    scatter_scores_kernel<<<(NC + 255) / 256, 256, 0, stream>>>(
        coords, params, bco, pose, res, atom, types, flat);
    gather_out_kernel<<<(NPAIR + 255) / 256, 256, 0, stream>>>(di, flat, out);
}